// EventSimulator_torch_39745627357266
// MI455X (gfx1250) — compile-verified
//
#include <hip/hip_runtime.h>
#include <hip/hip_bf16.h>
#include <stdint.h>

// ---------------------------------------------------------------------------
// ESIM event simulator for MI455X (gfx1250, wave32).
//   K1 gen_events   : TDM tensor_load_to_lds stages [256 pix x T frames] tile,
//                     per-pixel sequential reference scan emits (key,val) pairs.
//   K2..K5 per pass : 8-bit LSD radix sort (4 passes) of 18.43M 32-bit keys.
//                     Block scans use v_wmma_f32_16x16x4_f32 (prefix sum via
//                     triangular-matrix GEMM, exact for integers < 2^24).
//   Final scatter decodes payload and writes tf/xs/ys/ps/vf directly to d_out.
// ---------------------------------------------------------------------------

typedef float v2f __attribute__((ext_vector_type(2)));
typedef float v8f __attribute__((ext_vector_type(8)));
typedef unsigned int v4u __attribute__((ext_vector_type(4)));
typedef int v4i __attribute__((ext_vector_type(4)));
typedef int v8i __attribute__((ext_vector_type(8)));

#define IMG_H 480
#define IMG_W 640
#define IMG_HW (IMG_H * IMG_W)
#define CT_POSF 0.2f
#define CT_NEGF 0.2f
#define EPSF 1e-12f
#define SORT_BLK 2048  // keys per radix block (256 thr x 8 items)

// ---------------------------------------------------------------------------
// Kernel 1: event generation. One block = 256 consecutive pixels, all T frames
// staged into LDS by the Tensor Data Mover (2D descriptor: 256 x T tile,
// row stride = H*W floats).
// ---------------------------------------------------------------------------
__global__ void gen_events(const float* __restrict__ images,
                           const int* __restrict__ tstamps,
                           unsigned* __restrict__ keys,
                           unsigned* __restrict__ vals,
                           float* __restrict__ refOut, int T) {
  extern __shared__ float tile[];  // [T][256]
  const int t = threadIdx.x;
  const int p0 = blockIdx.x * 256;

  if ((t >> 5) == 0) {  // wave 0 issues the TDM descriptor (EXEC full in wave)
    unsigned long long ga = (unsigned long long)(uintptr_t)(images + p0);
    unsigned ldsOff = (unsigned)(uintptr_t)(&tile[0]);  // low 32 = LDS offset
    // D# group 0: count=1 | lds_addr | global_addr[56:0] | type=2
    v4u g0 = {1u, ldsOff, (unsigned)ga,
              (unsigned)((ga >> 32) & 0x1FFFFFFu) | (2u << 30)};
    // D# group 1: data_size=4B, tensor_dim0=256, tensor_dim1=T,
    // tile_dim0=256, tile_dim1=T, tensor_dim0_stride=IMG_HW
    v8i g1;
    g1[0] = (int)(2u << 16);                              // data_size=2 (4B)
    g1[1] = (int)((256u & 0xFFFFu) << 16);                // tensor_dim0 lo16
    g1[2] = (int)(((unsigned)T & 0xFFFFu) << 16);         // td0 hi16=0 | td1 lo16
    g1[3] = (int)(256u << 16);                            // td1 hi16=0 | tile_dim0
    g1[4] = (int)((unsigned)T & 0xFFFFu);                 // tile_dim1 | tile_dim2=0
    g1[5] = (int)(unsigned)(IMG_HW);                      // dim0_stride lo32
    g1[6] = 0;                                            // dim0_stride hi16 | d1s lo16
    g1[7] = 0;
    v4i z4 = {0, 0, 0, 0};              // groups 2/3 unused for 2D tensor
    v8i z8 = {0, 0, 0, 0, 0, 0, 0, 0};  // extra group (clang-23 6-arg form)
    __builtin_amdgcn_tensor_load_to_lds(g0, g1, z4, z4, z8, 0);
    __builtin_amdgcn_s_wait_tensorcnt(0);
  }
  __syncthreads();

  const int p = p0 + t;
  float ref = tile[t];  // frame 0
  for (int s = 0; s < T - 1; ++s) {
    const float it = tile[s * 256 + t];
    const float itdt = tile[(s + 1) * 256 + t];
    const float t0 = (float)tstamps[s];
    const float t1 = (float)tstamps[s + 1];
    const float d = itdt - ref;
    const bool pos = (d >= 0.f);
    const float pol = pos ? 1.f : -1.f;
    const float ct = pos ? CT_POSF : CT_NEGF;
    const float k = fminf(floorf(fabsf(d) / ct), 4.f);
    const float denom = itdt - it;
    const bool dok = fabsf(denom) > EPSF;
    const float safe = dok ? denom : 1.f;
    const float dt = t1 - t0;
    const unsigned base = ((unsigned)(s * IMG_HW + p)) << 2;
    uint4 K, V;
    unsigned kk[4], vv[4];
#pragma unroll
    for (int jj = 0; jj < 4; ++jj) {
      const float j = (float)(jj + 1);
      const float level = ref + pol * ct * j;
      const float frac = (level - it) / safe;
      const float tev = t0 + frac * dt;
      const bool valid = (j <= k) && dok && (tev > 0.f);
      // monotone float->uint (tev>0 when valid); invalid sorts last
      kk[jj] = valid ? (__float_as_uint(tev) | 0x80000000u) : 0xFFFFFFFFu;
      vv[jj] = (base + (unsigned)jj) | (pos ? (1u << 25) : 0u) |
               (valid ? (1u << 26) : 0u);
    }
    K.x = kk[0]; K.y = kk[1]; K.z = kk[2]; K.w = kk[3];
    V.x = vv[0]; V.y = vv[1]; V.z = vv[2]; V.w = vv[3];
    *(uint4*)(keys + base) = K;
    *(uint4*)(vals + base) = V;
    ref += pol * ct * k;
  }
  refOut[p] = ref;
}

__global__ void pad_init(unsigned* __restrict__ k, unsigned* __restrict__ v,
                         size_t N, size_t Npad) {
  for (size_t i = N + threadIdx.x; i < Npad; i += 256) {
    k[i] = 0xFFFFFFFFu;
    v[i] = 0u;
  }
}

// ---------------------------------------------------------------------------
// Radix sort, pass structure: hist -> scan_partial(WMMA) -> scan_sums ->
// scan_add -> scatter.
// ---------------------------------------------------------------------------
__global__ void hist_kernel(const unsigned* __restrict__ keys,
                            unsigned* __restrict__ counts, int nBlocks,
                            unsigned shift, size_t Npad) {
  __shared__ unsigned cnt[256];
  cnt[threadIdx.x] = 0u;
  __syncthreads();
  const size_t base = (size_t)blockIdx.x * SORT_BLK;
#pragma unroll
  for (int j = 0; j < 8; ++j) {
    const size_t g = base + (size_t)j * 256 + threadIdx.x;
    if (g < Npad) {
      const unsigned d = (keys[g] >> shift) & 255u;
      atomicAdd(&cnt[d], 1u);
    }
  }
  __syncthreads();
  // digit-major layout: counts[digit][block]
  counts[(size_t)threadIdx.x * (size_t)nBlocks + blockIdx.x] = cnt[threadIdx.x];
}

// Per-wave inclusive scan of 256 f32 values using v_wmma_f32_16x16x4_f32.
// X viewed 16x16 row-major; scan = X*U (row scans) + exclusive row-total scan.
// A layout (16x4 f32): lane m=l%16; reg j holds K = 4kk + 2*(l>=16) + j.
// B layout (4x16 f32): reg j holds row K = 4kk + 2j + (l>=16), col N = l%16.
// C/D layout: reg v holds row v + 8*(l>=16), col l%16.
__device__ __forceinline__ void wave_scan256(float* w) {
  const int lane = threadIdx.x & 31;
  const int half = lane >> 4;
  const int mrow = lane & 15;
  v2f A[4];
#pragma unroll
  for (int kk = 0; kk < 4; ++kk)
#pragma unroll
    for (int j = 0; j < 2; ++j)
      A[kk][j] = w[16 * mrow + 4 * kk + 2 * half + j];
  v8f S = {0.f, 0.f, 0.f, 0.f, 0.f, 0.f, 0.f, 0.f};  // X*U
  v8f R = {0.f, 0.f, 0.f, 0.f, 0.f, 0.f, 0.f, 0.f};  // X*J (row sums, all cols)
#pragma unroll
  for (int kk = 0; kk < 4; ++kk) {
    v2f BU, BJ;
#pragma unroll
    for (int j = 0; j < 2; ++j) {
      const int K = 4 * kk + 2 * j + half;
      BU[j] = (K <= mrow) ? 1.f : 0.f;  // upper-triangular ones
      BJ[j] = 1.f;                      // all ones
    }
    S = __builtin_amdgcn_wmma_f32_16x16x4_f32(false, A[kk], false, BU,
                                              (short)0, S, false, false);
    R = __builtin_amdgcn_wmma_f32_16x16x4_f32(false, A[kk], false, BJ,
                                              (short)0, R, false, false);
  }
  // exclusive scan of the 16 row totals: lane-local over its 8 rows, then
  // lanes 16..31 (rows 8..15) add total of rows 0..7 from lane (l&15).
  float e = 0.f;
  float outv[8];
#pragma unroll
  for (int v = 0; v < 8; ++v) {
    outv[v] = S[v] + e;
    e += R[v];
  }
  const float lower = __shfl(e, lane & 15, 32);
  const float add = half ? lower : 0.f;
#pragma unroll
  for (int v = 0; v < 8; ++v) w[16 * (v + 8 * half) + mrow] = outv[v] + add;
}

__global__ void scan_partial(const unsigned* __restrict__ counts,
                             unsigned* __restrict__ scanned,
                             unsigned* __restrict__ blockSums, size_t NC) {
  __shared__ float lds[SORT_BLK];
  const int t = threadIdx.x;
  const size_t base = (size_t)blockIdx.x * SORT_BLK;
#pragma unroll
  for (int j = 0; j < 8; ++j) {
    const size_t g = base + (size_t)j * 256 + t;
    lds[j * 256 + t] = (g < NC) ? (float)counts[g] : 0.f;
  }
  __syncthreads();
  wave_scan256(&lds[(t >> 5) * 256]);  // 8 waves, 256 elements each
  __syncthreads();
  float woff[8];
  float acc = 0.f;
#pragma unroll
  for (int wv = 0; wv < 8; ++wv) {
    woff[wv] = acc;
    acc += lds[wv * 256 + 255];  // each wave's inclusive total
  }
#pragma unroll
  for (int j = 0; j < 8; ++j) {
    const size_t g = base + (size_t)j * 256 + t;
    if (g < NC) scanned[g] = (unsigned)(lds[j * 256 + t] + woff[j] + 0.5f);
  }
  if (t == 0) blockSums[blockIdx.x] = (unsigned)(acc + 0.5f);
}

// Single-block integer scan of up to 2048 block sums (values can exceed 2^24,
// so this level stays exact uint32). Writes EXCLUSIVE scan in place.
__global__ void scan_sums(unsigned* __restrict__ blockSums, int n) {
  __shared__ unsigned s[2048];
  const int t = threadIdx.x;  // 1024 threads
  s[t] = (t < n) ? blockSums[t] : 0u;
  s[t + 1024] = (t + 1024 < n) ? blockSums[t + 1024] : 0u;
  __syncthreads();
  for (int off = 1; off < 2048; off <<= 1) {
    const unsigned a = (t >= off) ? s[t - off] : 0u;
    const unsigned b = (t + 1024 >= off) ? s[t + 1024 - off] : 0u;
    __syncthreads();
    s[t] += a;
    s[t + 1024] += b;
    __syncthreads();
  }
  if (t < n) blockSums[t] = (t == 0) ? 0u : s[t - 1];
  if (t + 1024 < n) blockSums[t + 1024] = s[t + 1023];
}

// scanned <- exclusive global offsets = incl_chunk - counts + chunk_base
__global__ void scan_add(unsigned* __restrict__ scanned,
                         const unsigned* __restrict__ counts,
                         const unsigned* __restrict__ blockSums, size_t NC) {
  const size_t i = (size_t)blockIdx.x * 256 + threadIdx.x;
  if (i < NC) scanned[i] = scanned[i] - counts[i] + blockSums[i >> 11];
}

// Stable scatter: per-iteration ranking with wave32 ballots + LDS wave counts.
__global__ void scatter_kernel(const unsigned* __restrict__ sk,
                               const unsigned* __restrict__ sv,
                               const unsigned* __restrict__ offs,
                               unsigned* __restrict__ dk,
                               unsigned* __restrict__ dv, int nBlocks,
                               unsigned shift, int finalPass,
                               float* __restrict__ out, size_t N) {
  __shared__ unsigned cnt[256];     // running block histogram
  __shared__ unsigned cw[8][256];   // per-wave counts this iteration
  const int t = threadIdx.x;
  const int lane = t & 31;
  const int wv = t >> 5;
  cnt[t] = 0u;
#pragma unroll
  for (int w = 0; w < 8; ++w) cw[w][t] = 0u;
  __syncthreads();
  const size_t base = (size_t)blockIdx.x * SORT_BLK;
#pragma unroll
  for (int it = 0; it < 8; ++it) {
    const size_t g = base + (size_t)it * 256 + t;  // Npad multiple of 2048
    const unsigned key = sk[g];
    const unsigned val = sv[g];
    const unsigned d = (key >> shift) & 255u;
    unsigned m = 0xFFFFFFFFu;  // lanes matching my digit
#pragma unroll
    for (int b = 0; b < 8; ++b) {
      const unsigned bal = __builtin_amdgcn_ballot_w32((d >> b) & 1u);
      m &= ((d >> b) & 1u) ? bal : ~bal;
    }
    const unsigned laneRank = __popc(m & ((1u << lane) - 1u));
    const unsigned baseD = cnt[d];  // pre-iteration running count
    if (laneRank == 0) cw[wv][d] = (unsigned)__popc(m);
    __syncthreads();
    unsigned prefW = 0;
    for (int w = 0; w < wv; ++w) prefW += cw[w][d];
    const size_t pos = (size_t)offs[(size_t)d * (size_t)nBlocks + blockIdx.x] +
                       baseD + prefW + laneRank;
    if (!finalPass) {
      dk[pos] = key;
      dv[pos] = val;
    } else if (pos < N) {
      const unsigned idx = val & 0x1FFFFFFu;
      const bool valid = (val >> 26) & 1u;
      const bool pp = (val >> 25) & 1u;
      const unsigned pix = (idx >> 2) % (unsigned)IMG_HW;
      const float tf =
          valid ? __uint_as_float(key & 0x7FFFFFFFu) : __builtin_inff();
      out[pos] = tf;
      out[N + pos] = (float)(pix % IMG_W);
      out[2 * N + pos] = (float)(pix / IMG_W);
      out[3 * N + pos] = pp ? 1.f : -1.f;
      out[4 * N + pos] = valid ? 1.f : 0.f;
    }
    __syncthreads();
    if (laneRank == 0) {
      atomicAdd(&cnt[d], (unsigned)__popc(m));
      cw[wv][d] = 0u;
    }
    __syncthreads();
  }
}

// ---------------------------------------------------------------------------
extern "C" void kernel_launch(void* const* d_in, const int* in_sizes, int n_in,
                              void* d_out, int out_size, void* d_ws,
                              size_t ws_size, hipStream_t stream) {
  const float* images = (const float*)d_in[0];
  const int* tstamps = (const int*)d_in[1];
  const int T = in_sizes[1];
  const size_t HW = (size_t)IMG_HW;
  const size_t N = (size_t)(T - 1) * HW * 4;
  const size_t Npad = (N + (SORT_BLK - 1)) & ~((size_t)SORT_BLK - 1);
  float* out = (float*)d_out;

  // carve workspace
  unsigned char* w = (unsigned char*)d_ws;
  auto carve = [&](size_t bytes) -> void* {
    void* p = (void*)w;
    w += (bytes + 255) & ~(size_t)255;
    return p;
  };
  unsigned* keysA = (unsigned*)carve(Npad * 4);
  unsigned* valsA = (unsigned*)carve(Npad * 4);
  unsigned* keysB = (unsigned*)carve(Npad * 4);
  unsigned* valsB = (unsigned*)carve(Npad * 4);
  const int nHB = (int)(Npad / SORT_BLK);
  const size_t NC = 256ull * (size_t)nHB;
  unsigned* counts = (unsigned*)carve(NC * 4);
  unsigned* scanned = (unsigned*)carve(NC * 4);
  const int nSB = (int)((NC + (SORT_BLK - 1)) / SORT_BLK);  // <= 2048 assumed
  unsigned* blockSums = (unsigned*)carve(2048 * 4);
  (void)ws_size;
  (void)n_in;
  (void)out_size;

  if (Npad > N) pad_init<<<1, 256, 0, stream>>>(keysA, valsA, N, Npad);
  gen_events<<<(unsigned)(HW / 256), 256, (size_t)T * 256 * sizeof(float),
               stream>>>(images, tstamps, keysA, valsA, out + 5 * N, T);

  unsigned* sk = keysA;
  unsigned* sv = valsA;
  unsigned* dk = keysB;
  unsigned* dv = valsB;
  for (int pass = 0; pass < 4; ++pass) {
    const unsigned shift = 8u * (unsigned)pass;
    hist_kernel<<<nHB, 256, 0, stream>>>(sk, counts, nHB, shift, Npad);
    scan_partial<<<nSB, 256, 0, stream>>>(counts, scanned, blockSums, NC);
    scan_sums<<<1, 1024, 0, stream>>>(blockSums, nSB);
    scan_add<<<(unsigned)((NC + 255) / 256), 256, 0, stream>>>(scanned, counts,
                                                               blockSums, NC);
    scatter_kernel<<<nHB, 256, 0, stream>>>(sk, sv, scanned, dk, dv, nHB, shift,
                                            (pass == 3) ? 1 : 0, out, N);
    unsigned* tk = sk; sk = dk; dk = tk;
    unsigned* tv = sv; sv = dv; dv = tv;
  }
}